// PairwisePotential_ori_25812753449052
// MI455X (gfx1250) — compile-verified
//
#include <hip/hip_runtime.h>
#include <hip/hip_bf16.h>
#include <math.h>

#define CHN 21
#define IMG 256
#define BATCH 8
#define NPIX (IMG * IMG)                    // 65536
#define NCE (BATCH * NPIX)                  // 524288
#define NTOT (BATCH * CHN * NPIX)           // 11010048
#define TILE 32
#define HALO 2
#define TDIM (TILE + 2 * HALO)              // 36
#define TELEMS (TDIM * TDIM)                // 1296
#define TILES_PER_PLANE ((IMG / TILE) * (IMG / TILE))  // 64
#define NPLANES (BATCH * CHN)               // 168
#define NPART (NPLANES * TILES_PER_PLANE)   // 10752 (divisible by 64)

typedef float f32x2 __attribute__((ext_vector_type(2)));
typedef float f32x8 __attribute__((ext_vector_type(8)));

#if defined(__AMDGCN__) && __has_builtin(__builtin_amdgcn_global_load_async_to_lds_b32)
#define ASYNC_LDS_OK 1
typedef __attribute__((address_space(1))) int* gptr_i32;
typedef __attribute__((address_space(3))) int* lptr_i32;
#else
#define ASYNC_LDS_OK 0
#endif

// ---------------------------------------------------------------------------
// Kernel 1: per-pixel cross entropy over 21 channels (register-resident).
// ---------------------------------------------------------------------------
__global__ __launch_bounds__(256) void ce_kernel(const float* __restrict__ x,
                                                 const int* __restrict__ lab,
                                                 float* __restrict__ ce) {
  const int idx = blockIdx.x * 256 + threadIdx.x;   // [0, NCE), exact grid
  const int b = idx >> 16;
  const int pix = idx & 0xFFFF;
  const float* xb = x + (size_t)b * CHN * NPIX + pix;
  const int l = lab[idx];

  float v[CHN];
  float m = -INFINITY;
  float xl = 0.0f;
#pragma unroll
  for (int c = 0; c < CHN; ++c) {
    float t = xb[(size_t)c * NPIX];
    v[c] = t;
    m = fmaxf(m, t);
    xl = (c == l) ? t : xl;   // select, keeps v[] in registers (no scratch)
  }
  float s = 0.0f;
#pragma unroll
  for (int c = 0; c < CHN; ++c) s += __expf(v[c] - m);
  ce[idx] = __logf(s) + m - xl;
}

// ---------------------------------------------------------------------------
// Kernel 2: pairwise gaussian potential for x and label on 32x32 tiles with a
// 2-wide halo staged in LDS (async-to-LDS path when available), accumulating
// (g - gl)^2 into one deterministic partial per block.
// ---------------------------------------------------------------------------
__global__ __launch_bounds__(256) void pot_kernel(const float* __restrict__ x,
                                                  const float* __restrict__ lab,
                                                  float* __restrict__ partial) {
  __shared__ float sx[TELEMS];
  __shared__ float sl[TELEMS];
  __shared__ float red[256];

  const int tid = threadIdx.x;
  const int plane = blockIdx.x / TILES_PER_PLANE;   // b*CHN + c
  const int tile = blockIdx.x % TILES_PER_PLANE;
  const int tx0 = (tile & 7) * TILE;
  const int ty0 = (tile >> 3) * TILE;
  const float* xp = x + (size_t)plane * NPIX;
  const float* lp = lab + (size_t)plane * NPIX;

  // Zero-fill (implements the reference's zero padding), then stage tile.
  for (int i = tid; i < TELEMS; i += 256) { sx[i] = 0.0f; sl[i] = 0.0f; }
  __syncthreads();

  for (int i = tid; i < TELEMS; i += 256) {
    const int r = i / TDIM;
    const int cix = i - r * TDIM;
    const int gy = ty0 + r - HALO;
    const int gx = tx0 + cix - HALO;
    if (gy >= 0 && gy < IMG && gx >= 0 && gx < IMG) {
      const unsigned gofs = (unsigned)(gy * IMG + gx);
#if ASYNC_LDS_OK
      __builtin_amdgcn_global_load_async_to_lds_b32(
          (gptr_i32)(uintptr_t)(xp + gofs),
          (lptr_i32)(uintptr_t)(unsigned)(uintptr_t)&sx[i],
          0, 0);
      __builtin_amdgcn_global_load_async_to_lds_b32(
          (gptr_i32)(uintptr_t)(lp + gofs),
          (lptr_i32)(uintptr_t)(unsigned)(uintptr_t)&sl[i],
          0, 0);
#else
      sx[i] = xp[gofs];
      sl[i] = lp[gofs];
#endif
    }
  }
#if ASYNC_LDS_OK
  asm volatile("s_wait_asynccnt 0" ::: "memory");
#endif
  __syncthreads();

  // 16 offsets: (dy,dx) with both components nonzero in [-2,2]; coef = -0.5/dist.
  constexpr int OY[16] = {-2, -2, -2, -2, -1, -1, -1, -1, 1, 1, 1, 1, 2, 2, 2, 2};
  constexpr int OX[16] = {-2, -1, 1, 2, -2, -1, 1, 2, -2, -1, 1, 2, -2, -1, 1, 2};
  constexpr float COEF[16] = {
      -0.17677670f, -0.22360680f, -0.22360680f, -0.17677670f,
      -0.22360680f, -0.35355339f, -0.35355339f, -0.22360680f,
      -0.22360680f, -0.35355339f, -0.35355339f, -0.22360680f,
      -0.17677670f, -0.22360680f, -0.22360680f, -0.17677670f};

  float acc = 0.0f;
#pragma unroll 1
  for (int p = tid; p < TILE * TILE; p += 256) {
    const int py = p >> 5;
    const int px = p & 31;
    const int ci = (py + HALO) * TDIM + (px + HALO);
    const float a = sx[ci];
    const float bb = sl[ci];
    float g = 0.0f, gl = 0.0f;
#pragma unroll
    for (int k = 0; k < 16; ++k) {
      const int ni = ci + OY[k] * TDIM + OX[k];
      const float d1 = a - sx[ni];
      const float d2 = bb - sl[ni];
      g += __expf(COEF[k] * d1 * d1);
      gl += __expf(COEF[k] * d2 * d2);
    }
    const float d = g - gl;
    acc += d * d;
  }

  // Deterministic block tree reduction.
  red[tid] = acc;
  __syncthreads();
  for (int s = 128; s > 0; s >>= 1) {
    if (tid < s) red[tid] += red[tid + s];
    __syncthreads();
  }
  if (tid == 0) partial[blockIdx.x] = red[0];
}

// ---------------------------------------------------------------------------
// Kernel 3: single-wave final reduction using V_WMMA_F32_16X16X4_F32 with an
// all-ones B matrix: D[m,n] = sum_k A[m,k] + C[m,n]. Each WMMA folds 64 f32
// partials (A layout is a bijection of the 64 loaded values onto (m,k) slots,
// and summation is permutation invariant). Column-n total = lane n + lane n+16
// of the per-lane sum over the 8 D VGPRs.
// ---------------------------------------------------------------------------
__global__ __launch_bounds__(32) void reduce_wmma_kernel(const float* __restrict__ partials,
                                                         float* __restrict__ out) {
  const int lane = threadIdx.x;       // 0..31, single wave, EXEC all ones
  f32x2 bones = {1.0f, 1.0f};
  f32x8 acc = {};
#pragma unroll 1
  for (int i = 0; i < NPART; i += 64) {
    f32x2 a;
    a.x = partials[i + lane];
    a.y = partials[i + 32 + lane];
    acc = __builtin_amdgcn_wmma_f32_16x16x4_f32(
        /*neg_a=*/false, a, /*neg_b=*/false, bones,
        /*c_mod=*/(short)0, acc, /*reuse_a=*/false, /*reuse_b=*/false);
  }
  float s = acc[0] + acc[1] + acc[2] + acc[3] + acc[4] + acc[5] + acc[6] + acc[7];
  const float tot = __shfl(s, 0, 32) + __shfl(s, 16, 32);
  if (lane == 0) out[0] = tot * (1.0f / (float)NTOT);
}

// ---------------------------------------------------------------------------
extern "C" void kernel_launch(void* const* d_in, const int* in_sizes, int n_in,
                              void* d_out, int out_size, void* d_ws, size_t ws_size,
                              hipStream_t stream) {
  const float* x = (const float*)d_in[0];        // [8,21,256,256] f32
  const int* label_single = (const int*)d_in[1]; // [8,256,256] i32
  const float* label = (const float*)d_in[2];    // [8,21,256,256] f32

  float* ce = (float*)d_out;                     // first 524288 floats
  float* mse = (float*)d_out + NCE;              // last float
  float* partials = (float*)d_ws;                // NPART floats of scratch

  ce_kernel<<<NCE / 256, 256, 0, stream>>>(x, label_single, ce);
  pot_kernel<<<NPART, 256, 0, stream>>>(x, label, partials);
  reduce_wmma_kernel<<<1, 32, 0, stream>>>(partials, mse);
}